// Zconv_27616639714004
// MI455X (gfx1250) — compile-verified
//
#include <hip/hip_runtime.h>

#define CCH   64
#define NBINS 32
#define NPTS  600000
#define NSP   32768
#define KIN   2048        // CCH*NBINS
#define HID   1024
#define SSEG  (NSP*NBINS) // 1048576
#define ZMIN  (-4.0f)
#define VZI   4.0f        // 1/0.25
#define EPS   1e-3f
#define LDA   36          // padded LDS row stride (floats): 144B = multiple of 16

typedef __attribute__((ext_vector_type(2))) float v2f;
typedef __attribute__((ext_vector_type(8))) float v8f;

__device__ __forceinline__ unsigned lds_off_u32(const void* p) {
  return (unsigned)(unsigned long long)p;   // low 32 bits of generic addr = LDS byte offset
}
__device__ __forceinline__ void async_ld_b128(unsigned ldsoff, unsigned gbyteoff, const void* base) {
  asm volatile("global_load_async_to_lds_b128 %0, %1, %2"
               :: "v"(ldsoff), "v"(gbyteoff), "s"(base) : "memory");
}
__device__ __forceinline__ void wait_async0() {
  asm volatile("s_wait_asynccnt 0" ::: "memory");
}

// ---------------- point stage ----------------
__global__ __launch_bounds__(256) void pts_stats(const float* __restrict__ pts,
                                                 const float* __restrict__ w0,
                                                 float* __restrict__ s0, float* __restrict__ q0) {
  const int c = threadIdx.x & 63, sub = threadIdx.x >> 6;
  float w[8];
#pragma unroll
  for (int k = 0; k < 8; ++k) w[k] = w0[c*8 + k];
  float s = 0.f, q = 0.f;
  for (int i = blockIdx.x*4 + sub; i < NPTS; i += gridDim.x*4) {
    const float* p = pts + (size_t)i*9;
    float h = 0.f;
#pragma unroll
    for (int k = 0; k < 8; ++k) h = fmaf(p[1+k], w[k], h);
    s += h; q += h*h;
  }
  __shared__ float sh[256];
  sh[threadIdx.x] = s; __syncthreads();
  if (threadIdx.x < 64) {
    float v = sh[threadIdx.x] + sh[threadIdx.x+64] + sh[threadIdx.x+128] + sh[threadIdx.x+192];
    atomicAdd(&s0[c], v);
  }
  __syncthreads();
  sh[threadIdx.x] = q; __syncthreads();
  if (threadIdx.x < 64) {
    float v = sh[threadIdx.x] + sh[threadIdx.x+64] + sh[threadIdx.x+128] + sh[threadIdx.x+192];
    atomicAdd(&q0[c], v);
  }
}

__global__ void bn0_final(const float* s, const float* q, const float* g, const float* b,
                          float* a0, float* c0) {
  int c = threadIdx.x;
  float m = s[c] / (float)NPTS;
  float v = q[c] / (float)NPTS - m*m;
  float aa = g[c] * rsqrtf(v + EPS);
  a0[c] = aa; c0[c] = b[c] - m*aa;
}

__global__ __launch_bounds__(256) void pts_scatter(const float* __restrict__ pts,
                                                   const int* __restrict__ pil,
                                                   const float* __restrict__ spf,
                                                   const float* __restrict__ w0,
                                                   const float* __restrict__ a0,
                                                   const float* __restrict__ c0,
                                                   float* __restrict__ sums,
                                                   float* __restrict__ cnt) {
  const int c = threadIdx.x & 63, sub = threadIdx.x >> 6;
  float w[8];
#pragma unroll
  for (int k = 0; k < 8; ++k) w[k] = w0[c*8 + k];
  const float aa = a0[c], cc = c0[c];
  for (int i = blockIdx.x*4 + sub; i < NPTS; i += gridDim.x*4) {
    const float* p = pts + (size_t)i*9;
    float h = 0.f;
#pragma unroll
    for (int k = 0; k < 8; ++k) h = fmaf(p[1+k], w[k], h);
    float v = fmaxf(fmaf(aa, h, cc), 0.f);
    int pid = pil[i];
    v += spf[(size_t)pid*CCH + c];
    int zb = (int)floorf((p[6] - ZMIN) * VZI);
    zb = min(max(zb, 0), NBINS - 1);
    int seg = pid*NBINS + zb;
    atomicAdd(&sums[(size_t)seg*CCH + c], v);
    if (c == 0) atomicAdd(&cnt[seg], 1.0f);
  }
}

// per pillar: occupancy mask + divide bin sums by counts (in place)
__global__ __launch_bounds__(256) void finalize_src(float* __restrict__ sums,
                                                    const float* __restrict__ cnt,
                                                    float* __restrict__ maskf,
                                                    float* __restrict__ nOcc) {
  const int p = blockIdx.x, tid = threadIdx.x;
  __shared__ float sInv[NBINS];
  __shared__ int sOcc;
  if (tid == 0) sOcc = 0;
  __syncthreads();
  if (tid < NBINS) {
    float cv = cnt[p*NBINS + tid];
    sInv[tid] = 1.0f / fmaxf(cv, 1.0f);
    if (cv > 0.f) atomicAdd(&sOcc, 1);
  }
  __syncthreads();
  float mf = (sOcc >= 2) ? 1.0f : 0.0f;
  if (tid == 0) { maskf[p] = mf; atomicAdd(nOcc, mf); }
  float* row = sums + (size_t)p*KIN;
#pragma unroll
  for (int j = 0; j < 8; ++j) {
    int e = tid + j*256;
    row[e] *= sInv[e >> 6];
  }
}

// ---------------- GEMM1: x1[32768,1024] = src[32768,2048] @ w1[1024,2048]^T ----------------
// 128x128 tile, BK=32, double-buffered async LDS staging.
__global__ __launch_bounds__(256) void gemm1_kernel(const float* __restrict__ A,
                                                    const float* __restrict__ B,
                                                    const float* __restrict__ maskf,
                                                    float* __restrict__ X1,
                                                    float* __restrict__ s1, float* __restrict__ q1) {
  __shared__ float sA[2][128*LDA];
  __shared__ float sB[2][128*LDA];
  __shared__ float sM[128];
  const int tid = threadIdx.x;
  const int mBase = blockIdx.y * 128, nBase = blockIdx.x * 128;
  if (tid < 128) sM[tid] = maskf[mBase + tid];
  const int wave = tid >> 5, lane = tid & 31;
  const int wm = wave & 3, wn = wave >> 2;         // 4x2 wave grid: 32 rows x 64 cols
  const int col = lane & 15, hi = lane >> 4;
  v8f acc[2][4];
#pragma unroll
  for (int i = 0; i < 2; ++i)
#pragma unroll
    for (int j = 0; j < 4; ++j) acc[i][j] = (v8f){0.f,0.f,0.f,0.f,0.f,0.f,0.f,0.f};

  const unsigned ldsA = lds_off_u32(sA), ldsB = lds_off_u32(sB);
  const unsigned bufBytes = (unsigned)(128*LDA*4);
  const int r = tid >> 3, q = tid & 7;             // e = tid + i*256 -> r += 32*i
  const unsigned lo = (unsigned)((r*LDA + q*4) * 4);

  auto issue = [&](int buf, int kk) {
    unsigned la = ldsA + (unsigned)buf*bufBytes + lo;
    unsigned lb = ldsB + (unsigned)buf*bufBytes + lo;
    unsigned ga = (unsigned)(((mBase + r)*KIN + kk + q*4) * 4);
    unsigned gb = (unsigned)(((nBase + r)*KIN + kk + q*4) * 4);
#pragma unroll
    for (int i = 0; i < 4; ++i) {                  // rows r, r+32, r+64, r+96
      async_ld_b128(la, ga, A);
      async_ld_b128(lb, gb, B);
      la += (unsigned)(32*LDA*4); lb += (unsigned)(32*LDA*4);
      ga += (unsigned)(32*KIN*4); gb += (unsigned)(32*KIN*4);
    }
  };

  issue(0, 0);                                     // prologue
  for (int kk = 0; kk < KIN; kk += 32) {
    const int cur = (kk >> 5) & 1;
    wait_async0();                                 // loads for buf[cur] (issued last iter) done
    __syncthreads();                               // all waves: loads visible, prev reads of buf[1-cur] done
    if (kk + 32 < KIN) issue(1 - cur, kk + 32);    // DMA for next tile overlaps WMMA below
    const float* cA = sA[cur];
    const float* cB = sB[cur];
#pragma unroll
    for (int k0 = 0; k0 < 32; k0 += 4) {
      const int kc = k0 + hi*2;
      v2f a[2], b[4];
#pragma unroll
      for (int tr = 0; tr < 2; ++tr)
        a[tr] = *(const v2f*)&cA[(wm*32 + tr*16 + col)*LDA + kc];
#pragma unroll
      for (int tc = 0; tc < 4; ++tc)
        b[tc] = *(const v2f*)&cB[(wn*64 + tc*16 + col)*LDA + kc];
#pragma unroll
      for (int tr = 0; tr < 2; ++tr)
#pragma unroll
        for (int tc = 0; tc < 4; ++tc)
          acc[tr][tc] = __builtin_amdgcn_wmma_f32_16x16x4_f32(
              false, a[tr], false, b[tc], (short)0, acc[tr][tc], false, false);
    }
  }
  __syncthreads();
  // epilogue: store x1, accumulate masked BN1 stats
#pragma unroll
  for (int tc = 0; tc < 4; ++tc) {
    const int nl = wn*64 + tc*16 + col;
    float ps = 0.f, pq = 0.f;
#pragma unroll
    for (int tr = 0; tr < 2; ++tr)
#pragma unroll
      for (int j = 0; j < 8; ++j) {
        int ml = wm*32 + tr*16 + hi*8 + j;
        float v = acc[tr][tc][j];
        X1[(size_t)(mBase + ml)*HID + nBase + nl] = v;
        float mf = sM[ml];
        ps = fmaf(v, mf, ps); pq = fmaf(v*v, mf, pq);
      }
    atomicAdd(&s1[nBase + nl], ps);
    atomicAdd(&q1[nBase + nl], pq);
  }
}

__global__ void bn_final(const float* s, const float* q, const float* g, const float* b,
                         const float* nOcc, float* a, float* c, int n) {
  int i = blockIdx.x*blockDim.x + threadIdx.x;
  if (i >= n) return;
  float cnt = fmaxf(nOcc[0], 1.0f);
  float m = s[i] / cnt, v = q[i] / cnt - m*m;
  float aa = g[i] * rsqrtf(v + EPS);
  a[i] = aa; c[i] = b[i] - m*aa;
}

__global__ __launch_bounds__(256) void bn_apply1(float* __restrict__ x1,
                                                 const float* __restrict__ a1,
                                                 const float* __restrict__ c1) {
  size_t i = (size_t)blockIdx.x*256 + threadIdx.x;   // one float4
  float4 v = ((float4*)x1)[i];
  int n0 = ((int)(i & 255)) * 4;
  v.x = fmaxf(fmaf(a1[n0+0], v.x, c1[n0+0]), 0.f);
  v.y = fmaxf(fmaf(a1[n0+1], v.y, c1[n0+1]), 0.f);
  v.z = fmaxf(fmaf(a1[n0+2], v.z, c1[n0+2]), 0.f);
  v.w = fmaxf(fmaf(a1[n0+3], v.w, c1[n0+3]), 0.f);
  ((float4*)x1)[i] = v;
}

// ---------------- GEMM2: x2[32768,64] = x1[32768,1024] @ w2[64,1024]^T ----------------
__global__ __launch_bounds__(256) void gemm2_kernel(const float* __restrict__ A,
                                                    const float* __restrict__ B,
                                                    const float* __restrict__ maskf,
                                                    float* __restrict__ X2,
                                                    float* __restrict__ s2, float* __restrict__ q2) {
  __shared__ float sA[2][128*LDA];
  __shared__ float sB[2][64*LDA];
  __shared__ float sM[128];
  const int tid = threadIdx.x;
  const int mBase = blockIdx.x * 128;
  if (tid < 128) sM[tid] = maskf[mBase + tid];
  const int wave = tid >> 5, lane = tid & 31;
  const int wm = wave & 3, wn = wave >> 2;         // 4x2 grid: 32 rows x 32 cols
  const int col = lane & 15, hi = lane >> 4;
  v8f acc[2][2];
#pragma unroll
  for (int i = 0; i < 2; ++i)
#pragma unroll
    for (int j = 0; j < 2; ++j) acc[i][j] = (v8f){0.f,0.f,0.f,0.f,0.f,0.f,0.f,0.f};

  const unsigned ldsA = lds_off_u32(sA), ldsB = lds_off_u32(sB);
  const unsigned bufA = (unsigned)(128*LDA*4), bufB = (unsigned)(64*LDA*4);
  const int r = tid >> 3, q = tid & 7;
  const unsigned lo = (unsigned)((r*LDA + q*4) * 4);

  auto issue = [&](int buf, int kk) {
    unsigned la = ldsA + (unsigned)buf*bufA + lo;
    unsigned ga = (unsigned)(((mBase + r)*HID + kk + q*4) * 4);
#pragma unroll
    for (int i = 0; i < 4; ++i) {                  // A rows r, r+32, r+64, r+96
      async_ld_b128(la, ga, A);
      la += (unsigned)(32*LDA*4); ga += (unsigned)(32*HID*4);
    }
    unsigned lb = ldsB + (unsigned)buf*bufB + lo;
    unsigned gb = (unsigned)((r*HID + kk + q*4) * 4);
#pragma unroll
    for (int i = 0; i < 2; ++i) {                  // B rows r, r+32
      async_ld_b128(lb, gb, B);
      lb += (unsigned)(32*LDA*4); gb += (unsigned)(32*HID*4);
    }
  };

  issue(0, 0);
  for (int kk = 0; kk < HID; kk += 32) {
    const int cur = (kk >> 5) & 1;
    wait_async0();
    __syncthreads();
    if (kk + 32 < HID) issue(1 - cur, kk + 32);
    const float* cA = sA[cur];
    const float* cB = sB[cur];
#pragma unroll
    for (int k0 = 0; k0 < 32; k0 += 4) {
      const int kc = k0 + hi*2;
      v2f a[2], b[2];
#pragma unroll
      for (int tr = 0; tr < 2; ++tr)
        a[tr] = *(const v2f*)&cA[(wm*32 + tr*16 + col)*LDA + kc];
#pragma unroll
      for (int tc = 0; tc < 2; ++tc)
        b[tc] = *(const v2f*)&cB[(wn*32 + tc*16 + col)*LDA + kc];
#pragma unroll
      for (int tr = 0; tr < 2; ++tr)
#pragma unroll
        for (int tc = 0; tc < 2; ++tc)
          acc[tr][tc] = __builtin_amdgcn_wmma_f32_16x16x4_f32(
              false, a[tr], false, b[tc], (short)0, acc[tr][tc], false, false);
    }
  }
  __syncthreads();
#pragma unroll
  for (int tc = 0; tc < 2; ++tc) {
    const int nl = wn*32 + tc*16 + col;
    float ps = 0.f, pq = 0.f;
#pragma unroll
    for (int tr = 0; tr < 2; ++tr)
#pragma unroll
      for (int j = 0; j < 8; ++j) {
        int ml = wm*32 + tr*16 + hi*8 + j;
        float v = acc[tr][tc][j];
        X2[(size_t)(mBase + ml)*CCH + nl] = v;
        float mf = sM[ml];
        ps = fmaf(v, mf, ps); pq = fmaf(v*v, mf, pq);
      }
    atomicAdd(&s2[nl], ps);
    atomicAdd(&q2[nl], pq);
  }
}

__global__ void final_out(const float* __restrict__ spf, const float* __restrict__ x2,
                          const float* __restrict__ a2, const float* __restrict__ c2,
                          const float* __restrict__ maskf, float* __restrict__ out) {
  int i = blockIdx.x*256 + threadIdx.x;     // 2M
  int c = i & (CCH-1), p = i >> 6;
  float v = fmaxf(fmaf(a2[c], x2[i], c2[c]), 0.f);
  out[i] = spf[i] + v * maskf[p];
}

extern "C" void kernel_launch(void* const* d_in, const int* in_sizes, int n_in,
                              void* d_out, int out_size, void* d_ws, size_t ws_size,
                              hipStream_t stream) {
  const float* pts = (const float*)d_in[0];
  const float* spf = (const float*)d_in[1];
  const float* w0  = (const float*)d_in[2];
  const float* g0  = (const float*)d_in[3];
  const float* b0  = (const float*)d_in[4];
  const float* w1  = (const float*)d_in[5];
  const float* g1  = (const float*)d_in[6];
  const float* b1  = (const float*)d_in[7];
  const float* w2  = (const float*)d_in[8];
  const float* g2  = (const float*)d_in[9];
  const float* b2  = (const float*)d_in[10];
  const int*   pil = (const int*)d_in[11];
  float* out = (float*)d_out;
  float* ws  = (float*)d_ws;

  float* sums  = ws;                               // [SSEG,64]  256MB
  float* cnt   = sums + (size_t)SSEG*CCH;          // [SSEG]       4MB
  float* x1    = cnt + SSEG;                       // [NSP,1024] 128MB
  float* x2    = x1 + (size_t)NSP*HID;             // [NSP,64]     8MB
  float* maskf = x2 + (size_t)NSP*CCH;             // [NSP]
  float* st    = maskf + NSP;
  float *s0 = st, *q0 = st+64, *a0 = st+128, *c0 = st+192;
  float *s1 = st+256, *q1 = st+1280, *a1 = st+2304, *c1 = st+3328;
  float *s2 = st+4352, *q2 = st+4416, *a2 = st+4480, *c2 = st+4544, *nOcc = st+4608;

  hipMemsetAsync(sums, 0, ((size_t)SSEG*CCH + SSEG)*sizeof(float), stream);
  hipMemsetAsync(st, 0, 4609*sizeof(float), stream);

  pts_stats<<<1024, 256, 0, stream>>>(pts, w0, s0, q0);
  bn0_final<<<1, 64, 0, stream>>>(s0, q0, g0, b0, a0, c0);
  pts_scatter<<<1024, 256, 0, stream>>>(pts, pil, spf, w0, a0, c0, sums, cnt);
  finalize_src<<<NSP, 256, 0, stream>>>(sums, cnt, maskf, nOcc);
  gemm1_kernel<<<dim3(8, 256), 256, 0, stream>>>(sums, w1, maskf, x1, s1, q1);
  bn_final<<<4, 256, 0, stream>>>(s1, q1, g1, b1, nOcc, a1, c1, HID);
  bn_apply1<<<(NSP*HID/4)/256, 256, 0, stream>>>(x1, a1, c1);
  gemm2_kernel<<<256, 256, 0, stream>>>(x1, w2, maskf, x2, s2, q2);
  bn_final<<<1, 64, 0, stream>>>(s2, q2, g2, b2, nOcc, a2, c2, CCH);
  final_out<<<NSP*CCH/256, 256, 0, stream>>>(spf, x2, a2, c2, maskf, out);
}